// pure_vit_decoder_5815385719055
// MI455X (gfx1250) — compile-verified
//
#include <hip/hip_runtime.h>
#include <hip/hip_bf16.h>

// ---------------------------------------------------------------------------
// pure_vit_decoder on MI455X (gfx1250):
//   ref: y = trilinear_upsample(x, 224, align_corners) ; z = 1x1x1conv(y,W,b)
//        out = softmax(z, dim=1)
// Linearity + (interp weights sum to 1) =>  conv-then-upsample is exact:
//        xc = W @ x + b   (tiny: 2744 voxels, 32->4 channels)  [WMMA f32]
//        out = softmax(trilinear(xc))                          [BW-bound]
// Roofline: output = 4*224^3 fp32 = 180 MB -> ~7.7us write floor @ 23.3 TB/s.
// ---------------------------------------------------------------------------

typedef __attribute__((ext_vector_type(2))) float v2f;
typedef __attribute__((ext_vector_type(8))) float v8f;

#define IN_D   14
#define IN_HW  (14 * 14)
#define IN_VOX (14 * 14 * 14)      // 2744 voxels
#define N_CH   32
#define N_CLS  4
#define OUTSZ  224
#define SCALE  (13.0f / 223.0f)    // align_corners: (in-1)/(out-1)

// ---------------------------------------------------------------------------
// Stage 1: xc[c][v] = sum_k W[c][k]*x[k][v] + b[c], via V_WMMA_F32_16X16X4_F32
// One wave per 16-voxel tile; K=32 covered by 8 chained K=4 WMMAs.
// All loads are unconditional (indices clamped, values masked with selects)
// so no exec-mask branching appears anywhere near the WMMA chain.
//
// ISA layouts (CDNA5 ISA):
//   A 16x4 f32 : lanes 0-15 -> M=lane, VGPR0=K0, VGPR1=K1
//                lanes 16-31 -> M=lane-16, VGPR0=K2, VGPR1=K3
//   B 4x16 f32 : mirrored (rows striped across lanes within a VGPR)
//   C/D 16x16  : VGPR j, lanes 0-15 -> M=j, N=lane; lanes 16-31 -> M=j+8
// ---------------------------------------------------------------------------
__global__ __launch_bounds__(32) void conv_fold_wmma(const float* __restrict__ x,
                                                     const float* __restrict__ W,
                                                     const float* __restrict__ bias,
                                                     float* __restrict__ xc) {
  const int lane = threadIdx.x;          // 0..31, full wave (EXEC all-1s)
  const int tile = blockIdx.x;           // 172 tiles of 16 voxels
  const int n    = lane & 15;            // N column / A row within half-wave
  const bool hi  = lane >= 16;
  const int koff = hi ? 2 : 0;           // K pair handled by this half-wave
  const int v    = tile * 16 + n;        // output voxel
  const int vc   = (v < IN_VOX) ? v : (IN_VOX - 1);  // clamped (pad lanes dropped)
  const bool arow = (n < N_CLS);         // real (non-padding) A row?
  const int  mrow = arow ? n : 0;        // clamped W row -> loads always in-bounds

  // --- Preload A (W slice) for all 8 K-steps: unconditional loads + cndmask.
  v2f a[8];
#pragma unroll
  for (int s = 0; s < 8; ++s) {
    const int k0 = 4 * s + koff;
    const float wx = W[mrow * N_CH + k0];
    const float wy = W[mrow * N_CH + k0 + 1];
    a[s].x = arow ? wx : 0.0f;
    a[s].y = arow ? wy : 0.0f;
  }

  // --- Preload B (x tile) for all 8 K-steps.
  v2f bm[8];
#pragma unroll
  for (int s = 0; s < 8; ++s) {
    const int k0 = 4 * s + koff;
    bm[s].x = x[(k0    ) * IN_VOX + vc];
    bm[s].y = x[(k0 + 1) * IN_VOX + vc];
  }

  // --- C init = bias broadcast along N (rows >= N_CLS are zero padding).
  v8f acc;
#pragma unroll
  for (int j = 0; j < 8; ++j) {
    const int row = j + (hi ? 8 : 0);
    const float bv = bias[(row < N_CLS) ? row : 0];
    acc[j] = (row < N_CLS) ? bv : 0.0f;
  }

  // --- 8 back-to-back WMMAs accumulate the K=32 contraction.
#pragma unroll
  for (int s = 0; s < 8; ++s) {
    // (neg_a, A, neg_b, B, c_mod, C, reuse_a, reuse_b)
    acc = __builtin_amdgcn_wmma_f32_16x16x4_f32(false, a[s], false, bm[s],
                                                (short)0, acc, false, false);
  }

  // Rows 0..3 live in acc[0..3] of lanes 0..15 (N = lane).
  if (!hi && v < IN_VOX) {
#pragma unroll
    for (int j = 0; j < N_CLS; ++j) xc[j * IN_VOX + v] = acc[j];
  }
}

// ---------------------------------------------------------------------------
// Stage 2: one block per output (d,h) row. Collapse d/h axes into a 4x14 LDS
// line (bilinear), then each of 224 threads lerps along w + softmax + 4
// coalesced stores. Write-bandwidth bound (~180 MB output).
// ---------------------------------------------------------------------------
__global__ __launch_bounds__(OUTSZ) void upsample_softmax(const float* __restrict__ xc,
                                                          float* __restrict__ out) {
  const int h = blockIdx.x;
  const int d = blockIdx.y;
  const int t = threadIdx.x;

  // align_corners source positions (uniform per block -> scalarized).
  float pd = d * SCALE; int d0 = (int)pd; if (d0 > IN_D - 2) d0 = IN_D - 2;
  float fd = pd - (float)d0;
  float ph = h * SCALE; int h0 = (int)ph; if (h0 > IN_D - 2) h0 = IN_D - 2;
  float fh = ph - (float)h0;

  __shared__ float line[N_CLS][16];      // 14 used, padded stride

  if (t < N_CLS * IN_D) {
    const int c  = t / IN_D;
    const int iw = t - c * IN_D;
    const float* base = xc + c * IN_VOX;
    const int i00 = (d0 * IN_D + h0) * IN_D + iw;
    const float v00 = base[i00];
    const float v01 = base[i00 + IN_D];            // h0+1
    const float v10 = base[i00 + IN_HW];           // d0+1
    const float v11 = base[i00 + IN_HW + IN_D];
    const float va = v00 + fh * (v01 - v00);
    const float vb = v10 + fh * (v11 - v10);
    line[c][iw] = va + fd * (vb - va);
  }
  __syncthreads();

  float pw = t * SCALE; int w0 = (int)pw; if (w0 > IN_D - 2) w0 = IN_D - 2;
  const float fw = pw - (float)w0;

  float z[N_CLS];
#pragma unroll
  for (int c = 0; c < N_CLS; ++c) {
    const float l0 = line[c][w0];
    const float l1 = line[c][w0 + 1];
    z[c] = l0 + fw * (l1 - l0);
  }

  // softmax over 4 classes
  const float mx = fmaxf(fmaxf(z[0], z[1]), fmaxf(z[2], z[3]));
  float e[N_CLS];
  float sum = 0.0f;
#pragma unroll
  for (int c = 0; c < N_CLS; ++c) {
    e[c] = __expf(z[c] - mx);
    sum += e[c];
  }
  const float inv = 1.0f / sum;

  const int base = (d * OUTSZ + h) * OUTSZ + t;    // [4][224][224][224]
#pragma unroll
  for (int c = 0; c < N_CLS; ++c)
    out[c * (OUTSZ * OUTSZ * OUTSZ) + base] = e[c] * inv;
}

// ---------------------------------------------------------------------------
extern "C" void kernel_launch(void* const* d_in, const int* in_sizes, int n_in,
                              void* d_out, int out_size, void* d_ws, size_t ws_size,
                              hipStream_t stream) {
  const float* x = (const float*)d_in[0];   // [1,32,14,14,14]
  const float* W = (const float*)d_in[1];   // [4,32]
  const float* b = (const float*)d_in[2];   // [4]
  float* xc  = (float*)d_ws;                // [4][2744] = 43,904 B scratch
  float* out = (float*)d_out;               // [1,4,224,224,224]

  conv_fold_wmma<<<dim3((IN_VOX + 15) / 16), dim3(32), 0, stream>>>(x, W, b, xc);
  upsample_softmax<<<dim3(OUTSZ, OUTSZ), dim3(OUTSZ), 0, stream>>>(xc, out);
}